// GuitarString_74440373174778
// MI455X (gfx1250) — compile-verified
//
#include <hip/hip_runtime.h>

// ---- CDNA5 async global<->LDS DMA via inline asm (gfx1250 only) ----
// The clang builtins on this toolchain want pointers in the cuda_device /
// cuda_shared language address spaces (unspellable in HIP source), so we issue
// the ISA instructions directly. Per CDNA5 ISA §15.18 / §10 (async fields):
//   global_load_async_to_lds_bN    vdst(=LDS addr VGPR), vaddr(64b), off
//   global_store_async_from_lds_bN vaddr(64b), vdata(=LDS addr VGPR), off
// Completion is tracked with ASYNCcnt -> s_wait_asynccnt.
#if defined(__gfx1250__)
#define GS_ASM_ASYNC 1
#else
#define GS_ASM_ASYNC 0
#endif

#if GS_ASM_ASYNC
__device__ __forceinline__ unsigned gs_lds_off(const void* p) {
  // Generic pointers into LDS carry the LDS byte offset in the low 32 bits
  // (ISA §10.2).
  return (unsigned)(unsigned long long)p;
}
__device__ __forceinline__ void gs_async_load_b128(const float* gsrc, unsigned ldsDst) {
  asm volatile("global_load_async_to_lds_b128 %0, %1, off"
               :
               : "v"(ldsDst), "v"((unsigned long long)(const void*)gsrc)
               : "memory");
}
__device__ __forceinline__ void gs_async_load_b32(const float* gsrc, unsigned ldsDst) {
  asm volatile("global_load_async_to_lds_b32 %0, %1, off"
               :
               : "v"(ldsDst), "v"((unsigned long long)(const void*)gsrc)
               : "memory");
}
__device__ __forceinline__ void gs_async_store_b128(float* gdst, unsigned ldsSrc) {
  asm volatile("global_store_async_from_lds_b128 %0, %1, off"
               :
               : "v"((unsigned long long)(void*)gdst), "v"(ldsSrc)
               : "memory");
}
__device__ __forceinline__ void gs_async_store_b32(float* gdst, unsigned ldsSrc) {
  asm volatile("global_store_async_from_lds_b32 %0, %1, off"
               :
               : "v"((unsigned long long)(void*)gdst), "v"(ldsSrc)
               : "memory");
}
__device__ __forceinline__ void gs_wait_async0() {
  asm volatile("s_wait_asynccnt 0x0" ::: "memory");
}
#endif

static constexpr float kNutGain    = -0.99f;
static constexpr float kBridgeGain = -0.99f;
static constexpr float kVib        = 0.999f;
static constexpr float kDispA      = 0.7f;

// One block per batch row. L/R delay lines resident in LDS (2*T*4 = 256 KB of
// the 320 KB/WGP). 1024 threads = 32 wave32 waves. Per-thread register
// accumulators over 32 strided time slots.
__global__ __launch_bounds__(1024)
void GuitarString_scan_kernel(const int*   __restrict__ lenp,
                              const float* __restrict__ pluckp,
                              const float* __restrict__ exc,
                              float*       __restrict__ out,
                              int T, int BT)
{
  constexpr int BS    = 1024;
  constexpr int ITEMS = 32;            // BS*ITEMS = 32768 >= T = 32000
  extern __shared__ float smem[];
  float* Lb = smem;                    // left-going wave (full row)
  float* Rb = smem + T;                // right-going wave (full row)

  const int tid = (int)threadIdx.x;
  const int b   = (int)blockIdx.x;

  // Scalar parameters (uniform; scalar loads)
  const int   length  = lenp[0];
  const int   n_loops = T / length;
  const float p   = 0.1f + 0.8f * pluckp[0];
  const float nUp = (float)length * p;
  const float nDn = (float)length - nUp;
  // frac_delay(x,d)[t] = (1-f)*x[t-K] + f*x[t-K+1], K=ceil(d), f=K-d, x[<0]=0
  const int   Ku  = (int)ceilf(nUp);
  const int   Kd  = (int)ceilf(nDn);
  const float fu  = (float)Ku - nUp;
  const float fd  = (float)Kd - nDn;
  const float Gr  = kNutGain * kVib;           // r_next gain
  const float Gl  = kBridgeGain * kVib * 0.5f; // l_next gain (incl. 0.5)

  const float* ex = exc + (size_t)b * (size_t)T;

  // ---- Stage excitation row (128 KB) into LDS via async DMA ----
#if GS_ASM_ASYNC
  {
    const unsigned ldsL = gs_lds_off(Lb);
    const int nv4 = T >> 2;
    for (int n = tid; n < nv4; n += BS)
      gs_async_load_b128(ex + 4 * n, ldsL + 16u * (unsigned)n);
    for (int n = (nv4 << 2) + tid; n < T; n += BS)   // tail (T % 4)
      gs_async_load_b32(ex + n, ldsL + 4u * (unsigned)n);
    gs_wait_async0();
  }
#else
  for (int n = tid; n < T; n += BS) Lb[n] = ex[n];
#endif
  __syncthreads();

  // ---- Init: l0 = r0 = 0.5*exc; accumulators start at l0/r0 ----
  float accL[ITEMS], accR[ITEMS];
#pragma unroll
  for (int i = 0; i < ITEMS; ++i) {
    int t = tid + i * BS;
    if (t < T) {
      float v = 0.5f * Lb[t];
      accL[i] = v; accR[i] = v;
      Lb[t] = v;  Rb[t] = v;          // own slot only: no cross-thread hazard
    } else { accL[i] = 0.0f; accR[i] = 0.0f; }
  }
  __syncthreads();

  // ---- 250 sequential waveguide steps, all state in LDS ----
  for (int s = 0; s < n_loops; ++s) {
    float nl[ITEMS], nr[ITEMS];
#pragma unroll
    for (int i = 0; i < ITEMS; ++i) {
      int t = tid + i * BS;
      if (t < T) {
        // r_next[t] = Gr * frac_delay(L, nUp)[t]
        int j = t - Ku;
        float a0 = (j     >= 0) ? Lb[j]     : 0.0f;
        float a1 = (j + 1 >= 0) ? Lb[j + 1] : 0.0f;
        nr[i] = Gr * ((1.0f - fu) * a0 + fu * a1);

        // l_next[t] = Gl * [(1-fd)*dr(k-1) + dr(k) + fd*dr(k+1)], k = t-Kd
        // dr(j) = 0.7*R[j] + 0.3*R[j-1], R[<0] = 0
        int k = t - Kd;
        float r0 = (k - 2 >= 0) ? Rb[k - 2] : 0.0f;
        float r1 = (k - 1 >= 0) ? Rb[k - 1] : 0.0f;
        float r2 = (k     >= 0) ? Rb[k]     : 0.0f;
        float r3 = (k + 1 >= 0) ? Rb[k + 1] : 0.0f;
        float d0 = kDispA * r1 + (1.0f - kDispA) * r0;
        float d1 = kDispA * r2 + (1.0f - kDispA) * r1;
        float d2 = kDispA * r3 + (1.0f - kDispA) * r2;
        nl[i] = Gl * ((1.0f - fd) * d0 + d1 + fd * d2);
      } else { nl[i] = 0.0f; nr[i] = 0.0f; }
    }
    __syncthreads();                   // all old-state reads done
#pragma unroll
    for (int i = 0; i < ITEMS; ++i) {
      int t = tid + i * BS;
      if (t < T) {
        Lb[t] = nl[i];  Rb[t] = nr[i];
        accL[i] += nl[i]; accR[i] += nr[i];
      }
    }
    __syncthreads();                   // new state visible for next step
  }

  // ---- Stage sums through LDS, async-store to global ----
#pragma unroll
  for (int i = 0; i < ITEMS; ++i) {
    int t = tid + i * BS;
    if (t < T) { Lb[t] = accL[i]; Rb[t] = accR[i]; }
  }
  __syncthreads();

  float* outL = out + (size_t)b * (size_t)T;              // ls
  float* outR = out + (size_t)BT + (size_t)b * (size_t)T; // rs
#if GS_ASM_ASYNC
  {
    const unsigned offL = gs_lds_off(Lb);
    const unsigned offR = gs_lds_off(Rb);
    const int nv4 = T >> 2;
    for (int n = tid; n < nv4; n += BS) {
      gs_async_store_b128(outL + 4 * n, offL + 16u * (unsigned)n);
      gs_async_store_b128(outR + 4 * n, offR + 16u * (unsigned)n);
    }
    for (int n = (nv4 << 2) + tid; n < T; n += BS) {
      gs_async_store_b32(outL + n, offL + 4u * (unsigned)n);
      gs_async_store_b32(outR + n, offR + 4u * (unsigned)n);
    }
    gs_wait_async0();                  // LDS must stay live until DMA done
  }
#else
  for (int n = tid; n < T; n += BS) { outL[n] = Lb[n]; outR[n] = Rb[n]; }
#endif
}

extern "C" void kernel_launch(void* const* d_in, const int* in_sizes, int n_in,
                              void* d_out, int out_size, void* d_ws, size_t ws_size,
                              hipStream_t stream) {
  const int*   lenp  = (const int*)  d_in[0];   // length (scalar int)
  const float* pluck = (const float*)d_in[1];   // pluckPosition (1 float)
  const float* exc   = (const float*)d_in[2];   // excitation (B*T f32)
  float*       out   = (float*)d_out;           // [ls (B*T) | rs (B*T)]

  const int BT = in_sizes[2];                   // 64 * 32000
  const int B  = 64;                            // fixed by reference setup
  const int T  = BT / B;                        // 32000

  const size_t lds_bytes = 2u * (size_t)T * sizeof(float); // 256 KB / 320 KB WGP
  GuitarString_scan_kernel<<<dim3(B), dim3(1024), lds_bytes, stream>>>(
      lenp, pluck, exc, out, T, BT);

  (void)n_in; (void)out_size; (void)d_ws; (void)ws_size;
}